// GNNFeatureExtractor_7919919693917
// MI455X (gfx1250) — compile-verified
//
#include <hip/hip_runtime.h>
#include <hip/hip_bf16.h>

typedef float v2f __attribute__((ext_vector_type(2)));
typedef float v8f __attribute__((ext_vector_type(8)));

#define HEADS 4
#define CH 64

// ---------------------------------------------------------------------------
// ordered-uint encoding for float atomic max
// ---------------------------------------------------------------------------
__device__ __forceinline__ unsigned enc_ord(float f) {
    unsigned u = __float_as_uint(f);
    return (u & 0x80000000u) ? ~u : (u | 0x80000000u);
}
__device__ __forceinline__ float dec_ord(unsigned u) {
    u = (u & 0x80000000u) ? (u ^ 0x80000000u) : ~u;
    return __uint_as_float(u);
}

__device__ __forceinline__ float wave_sum32(float v) {
    for (int o = 16; o > 0; o >>= 1) v += __shfl_xor(v, o, 32);
    return v;
}

// ---------------------------------------------------------------------------
// Node encoder: x = LN(relu([pos,deg] @ enc_w + enc_b)) ; one wave per node
// ---------------------------------------------------------------------------
__global__ void node_encoder_kernel(const float* __restrict__ pos,
                                    const float* __restrict__ deg,
                                    const float* __restrict__ enc_w,
                                    const float* __restrict__ enc_b,
                                    const float* __restrict__ enc_g,
                                    const float* __restrict__ enc_be,
                                    float* __restrict__ x, int n) {
    int node = (blockIdx.x * blockDim.x + threadIdx.x) >> 5;
    int lane = threadIdx.x & 31;
    if (node >= n) return;
    float f0 = pos[node * 3 + 0];
    float f1 = pos[node * 3 + 1];
    float f2 = pos[node * 3 + 2];
    float f3 = deg[node];
    float v = f0 * enc_w[0 * 32 + lane] + f1 * enc_w[1 * 32 + lane] +
              f2 * enc_w[2 * 32 + lane] + f3 * enc_w[3 * 32 + lane] + enc_b[lane];
    v = fmaxf(v, 0.0f);
    float mean = wave_sum32(v) * (1.0f / 32.0f);
    float dv = v - mean;
    float var = wave_sum32(dv * dv) * (1.0f / 32.0f);
    float xn = dv * rsqrtf(var + 1e-5f) * enc_g[lane] + enc_be[lane];
    x[node * 32 + lane] = xn;
}

// ---------------------------------------------------------------------------
// fp32 WMMA GEMM: C[M,Nc] = A[M,K] @ B[K,Nc].  One wave per 16x16 tile.
// Uses V_WMMA_F32_16X16X4_F32 (wave32).  M multiple of 16, K multiple of 4,
// Nc multiple of 16 for our shapes.
// ---------------------------------------------------------------------------
__global__ void wmma_gemm_f32_kernel(const float* __restrict__ A,
                                     const float* __restrict__ B,
                                     float* __restrict__ C,
                                     int M, int K, int Nc) {
    int wave = (blockIdx.x * blockDim.x + threadIdx.x) >> 5;
    int lane = threadIdx.x & 31;
    int tilesN = Nc >> 4;
    int tileM = wave / tilesN;
    int tileN = wave - tileM * tilesN;
    if (tileM * 16 >= M) return;   // uniform per wave -> EXEC stays all-ones

    int half = lane >> 4;          // 0: lanes 0-15, 1: lanes 16-31
    int mr = tileM * 16 + (lane & 15);
    int nc = tileN * 16 + (lane & 15);

    v8f acc = {0.f, 0.f, 0.f, 0.f, 0.f, 0.f, 0.f, 0.f};
    for (int k0 = 0; k0 < K; k0 += 4) {
        int ka = k0 + 2 * half;
        v2f a, b;
        a.x = A[(size_t)mr * K + ka];
        a.y = A[(size_t)mr * K + ka + 1];
        b.x = B[(size_t)ka * Nc + nc];
        b.y = B[(size_t)(ka + 1) * Nc + nc];
        acc = __builtin_amdgcn_wmma_f32_16x16x4_f32(false, a, false, b,
                                                    (short)0, acc, false, false);
    }
    for (int r = 0; r < 8; ++r) {
        int mm = tileM * 16 + r + 8 * half;
        C[(size_t)mm * Nc + nc] = acc[r];
    }
}

// ---------------------------------------------------------------------------
// a_s / a_d: per (node, head) dot over 64 channels
// ---------------------------------------------------------------------------
__global__ void attn_coef_kernel(const float* __restrict__ h,
                                 const float* __restrict__ att_s,
                                 const float* __restrict__ att_d,
                                 float* __restrict__ as_, float* __restrict__ ad_,
                                 int n, int heads) {
    int total = n * heads;
    for (int idx = blockIdx.x * blockDim.x + threadIdx.x; idx < total;
         idx += gridDim.x * blockDim.x) {
        int node = idx / heads;
        int hh = idx - node * heads;
        const float* hp = h + ((size_t)node * heads + hh) * CH;
        const float* sp = att_s + hh * CH;
        const float* dp = att_d + hh * CH;
        float ss = 0.f, dd = 0.f;
        for (int c = 0; c < CH; ++c) {
            float hv = hp[c];
            ss += hv * sp[c];
            dd += hv * dp[c];
        }
        as_[idx] = ss;
        ad_[idx] = dd;
    }
}

// ---------------------------------------------------------------------------
// edge pass 1: leaky-relu logits + segment max (ordered-uint atomicMax)
// ---------------------------------------------------------------------------
__global__ void edge_max_kernel(const int* __restrict__ ei,
                                const float* __restrict__ as_,
                                const float* __restrict__ ad_,
                                unsigned* __restrict__ emax,
                                int E_, int n, int heads) {
    int total = (E_ + n) * heads;
    for (int idx = blockIdx.x * blockDim.x + threadIdx.x; idx < total;
         idx += gridDim.x * blockDim.x) {
        int e = idx / heads;
        int hh = idx - e * heads;
        int s, d;
        if (e < E_) { s = ei[e]; d = ei[E_ + e]; }
        else        { s = d = e - E_; }
        float ev = as_[s * heads + hh] + ad_[d * heads + hh];
        ev = ev > 0.f ? ev : 0.2f * ev;
        atomicMax(&emax[d * heads + hh], enc_ord(ev));
    }
}

// ---------------------------------------------------------------------------
// edge pass 2: ex = exp(e - max) ; denom += ex ; store numerator
// ---------------------------------------------------------------------------
__global__ void edge_exp_denom_kernel(const int* __restrict__ ei,
                                      const float* __restrict__ as_,
                                      const float* __restrict__ ad_,
                                      const unsigned* __restrict__ emax,
                                      float* __restrict__ alpha_num,
                                      float* __restrict__ denom,
                                      int E_, int n, int heads) {
    int total = (E_ + n) * heads;
    for (int idx = blockIdx.x * blockDim.x + threadIdx.x; idx < total;
         idx += gridDim.x * blockDim.x) {
        int e = idx / heads;
        int hh = idx - e * heads;
        int s, d;
        if (e < E_) { s = ei[e]; d = ei[E_ + e]; }
        else        { s = d = e - E_; }
        float ev = as_[s * heads + hh] + ad_[d * heads + hh];
        ev = ev > 0.f ? ev : 0.2f * ev;
        float ex = expf(ev - dec_ord(emax[d * heads + hh]));
        alpha_num[idx] = ex;
        atomicAdd(&denom[d * heads + hh], ex);
    }
}

// ---------------------------------------------------------------------------
// edge pass 3: out[dst] += alpha * h[src]  (64 threads per edge)
// ---------------------------------------------------------------------------
__global__ void edge_aggregate_kernel(const int* __restrict__ ei,
                                      const float* __restrict__ alpha_num,
                                      const float* __restrict__ denom,
                                      const float* __restrict__ hsrc,
                                      float* __restrict__ out,
                                      int E_, int n, int heads) {
    long long total = (long long)(E_ + n) * CH;
    for (long long idx = (long long)blockIdx.x * blockDim.x + threadIdx.x; idx < total;
         idx += (long long)gridDim.x * blockDim.x) {
        int e = (int)(idx >> 6);
        int c = (int)(idx & 63);
        int s, d;
        if (e < E_) { s = ei[e]; d = ei[E_ + e]; }
        else        { s = d = e - E_; }
        for (int hh = 0; hh < heads; ++hh) {
            float alpha = alpha_num[(size_t)e * heads + hh] /
                          (denom[d * heads + hh] + 1e-16f);
            atomicAdd(&out[((size_t)d * heads + hh) * CH + c],
                      alpha * hsrc[((size_t)s * heads + hh) * CH + c]);
        }
    }
}

// ---------------------------------------------------------------------------
// post GAT1: h = elu(LN(out1 + gat1_b)) ; block(256) per node
// ---------------------------------------------------------------------------
__global__ void post1_kernel(const float* __restrict__ out1,
                             const float* __restrict__ bias,
                             const float* __restrict__ g,
                             const float* __restrict__ be,
                             float* __restrict__ hpost, int n) {
    __shared__ float red[256];
    int node = blockIdx.x;
    int t = threadIdx.x;
    float v = out1[(size_t)node * 256 + t] + bias[t];
    red[t] = v; __syncthreads();
    for (int s = 128; s > 0; s >>= 1) { if (t < s) red[t] += red[t + s]; __syncthreads(); }
    float mean = red[0] * (1.0f / 256.0f); __syncthreads();
    float dv = v - mean;
    red[t] = dv * dv; __syncthreads();
    for (int s = 128; s > 0; s >>= 1) { if (t < s) red[t] += red[t + s]; __syncthreads(); }
    float var = red[0] * (1.0f / 256.0f);
    float xn = dv * rsqrtf(var + 1e-5f) * g[t] + be[t];
    hpost[(size_t)node * 256 + t] = xn > 0.f ? xn : expm1f(xn);
}

// ---------------------------------------------------------------------------
// post GAT2: h = elu(LN(out2 + gat2_b) + identity + proj_b) ; pooled += h
// block(64) per node
// ---------------------------------------------------------------------------
__global__ void post2_kernel(const float* __restrict__ out2,
                             const float* __restrict__ bias,
                             const float* __restrict__ g,
                             const float* __restrict__ be,
                             const float* __restrict__ identW,
                             const float* __restrict__ proj_b,
                             float* __restrict__ pooled, int n) {
    __shared__ float red[64];
    int node = blockIdx.x;
    int t = threadIdx.x;
    float v = out2[(size_t)node * 64 + t] + bias[t];
    red[t] = v; __syncthreads();
    for (int s = 32; s > 0; s >>= 1) { if (t < s) red[t] += red[t + s]; __syncthreads(); }
    float mean = red[0] * (1.0f / 64.0f); __syncthreads();
    float dv = v - mean;
    red[t] = dv * dv; __syncthreads();
    for (int s = 32; s > 0; s >>= 1) { if (t < s) red[t] += red[t + s]; __syncthreads(); }
    float var = red[0] * (1.0f / 64.0f);
    float xn = dv * rsqrtf(var + 1e-5f) * g[t] + be[t];
    xn += identW[(size_t)node * 64 + t] + proj_b[t];
    float h = xn > 0.f ? xn : expm1f(xn);
    atomicAdd(&pooled[t], h);
}

// ---------------------------------------------------------------------------
// Fusion head: single block of 256 threads.
// comb = [pooled/N (64), LN(relu(traffic @ tr_w + tr_b)) (32)]
// out  = LN(relu(comb @ fu_w + fu_b))
// ---------------------------------------------------------------------------
__global__ void fusion_kernel(const float* __restrict__ traffic,
                              const float* __restrict__ tr_w,
                              const float* __restrict__ tr_b,
                              const float* __restrict__ tr_g,
                              const float* __restrict__ tr_be,
                              const float* __restrict__ pooled,
                              const float* __restrict__ fu_w,
                              const float* __restrict__ fu_b,
                              const float* __restrict__ fu_g,
                              const float* __restrict__ fu_be,
                              float* __restrict__ out, int n) {
    __shared__ float comb[96];
    __shared__ float tfs[32];
    __shared__ float red[256];
    int t = threadIdx.x;
    if (t < 32) {
        float tv = tr_b[t];
        for (int k = 0; k < 5; ++k) tv += traffic[k] * tr_w[k * 32 + t];
        tfs[t] = fmaxf(tv, 0.f);
    }
    if (t < 64) comb[t] = pooled[t] / (float)n;
    __syncthreads();
    if (t < 32) {
        float m = 0.f;
        for (int k = 0; k < 32; ++k) m += tfs[k];
        m *= (1.0f / 32.0f);
        float var = 0.f;
        for (int k = 0; k < 32; ++k) { float d = tfs[k] - m; var += d * d; }
        var *= (1.0f / 32.0f);
        comb[64 + t] = (tfs[t] - m) * rsqrtf(var + 1e-5f) * tr_g[t] + tr_be[t];
    }
    __syncthreads();
    float o = fu_b[t];
    for (int k = 0; k < 96; ++k) o += comb[k] * fu_w[k * 256 + t];
    o = fmaxf(o, 0.f);
    red[t] = o; __syncthreads();
    for (int s = 128; s > 0; s >>= 1) { if (t < s) red[t] += red[t + s]; __syncthreads(); }
    float mean = red[0] * (1.0f / 256.0f); __syncthreads();
    float dv = o - mean;
    red[t] = dv * dv; __syncthreads();
    for (int s = 128; s > 0; s >>= 1) { if (t < s) red[t] += red[t + s]; __syncthreads(); }
    float var = red[0] * (1.0f / 256.0f);
    out[t] = dv * rsqrtf(var + 1e-5f) * fu_g[t] + fu_be[t];
}

// ---------------------------------------------------------------------------
// launch
// ---------------------------------------------------------------------------
extern "C" void kernel_launch(void* const* d_in, const int* in_sizes, int n_in,
                              void* d_out, int out_size, void* d_ws, size_t ws_size,
                              hipStream_t stream) {
    const float* positions = (const float*)d_in[0];
    const float* degrees   = (const float*)d_in[1];
    const float* traffic   = (const float*)d_in[2];
    const int*   edge_idx  = (const int*)d_in[3];
    const float* enc_w  = (const float*)d_in[4];
    const float* enc_b  = (const float*)d_in[5];
    const float* enc_g  = (const float*)d_in[6];
    const float* enc_be = (const float*)d_in[7];
    const float* gat1_w  = (const float*)d_in[8];
    const float* gat1_as = (const float*)d_in[9];
    const float* gat1_ad = (const float*)d_in[10];
    const float* gat1_b  = (const float*)d_in[11];
    const float* n1_g = (const float*)d_in[12];
    const float* n1_b = (const float*)d_in[13];
    const float* proj_w = (const float*)d_in[14];
    const float* proj_b = (const float*)d_in[15];
    const float* gat2_w  = (const float*)d_in[16];
    const float* gat2_as = (const float*)d_in[17];
    const float* gat2_ad = (const float*)d_in[18];
    const float* gat2_b  = (const float*)d_in[19];
    const float* n2_g = (const float*)d_in[20];
    const float* n2_b = (const float*)d_in[21];
    const float* tr_w  = (const float*)d_in[22];
    const float* tr_b  = (const float*)d_in[23];
    const float* tr_g  = (const float*)d_in[24];
    const float* tr_be = (const float*)d_in[25];
    const float* fu_w  = (const float*)d_in[26];
    const float* fu_b  = (const float*)d_in[27];
    const float* fu_g  = (const float*)d_in[28];
    const float* fu_be = (const float*)d_in[29];

    const int N  = in_sizes[1];          // degrees is [N]
    const int E_ = in_sizes[3] / 2;      // edge_index is [2,E]
    const int EP = E_ + N;               // with self loops

    float* ws = (float*)d_ws;
    size_t off = 0;
    auto alloc = [&](size_t cnt) -> float* {
        float* p = ws + off;
        off += (cnt + 63) & ~(size_t)63;
        return p;
    };
    float*    x      = alloc((size_t)N * 32);
    float*    identW = alloc((size_t)N * 64);
    float*    h1     = alloc((size_t)N * 256);   // reused as hpost1 after agg
    float*    as1    = alloc((size_t)N * HEADS);
    float*    ad1    = alloc((size_t)N * HEADS);
    unsigned* emax1  = (unsigned*)alloc((size_t)N * HEADS);
    float*    denom1 = alloc((size_t)N * HEADS);
    float*    alpha1 = alloc((size_t)EP * HEADS);
    float*    out1   = alloc((size_t)N * 256);
    float*    h2     = alloc((size_t)N * 64);
    float*    as2    = alloc((size_t)N);
    float*    ad2    = alloc((size_t)N);
    unsigned* emax2  = (unsigned*)alloc((size_t)N);
    float*    denom2 = alloc((size_t)N);
    float*    alpha2 = alloc((size_t)EP);
    float*    out2   = alloc((size_t)N * 64);
    float*    pooled = alloc((size_t)64);

    // zero accumulators (encoded -inf for ordered-uint max buffers is > 0)
    hipMemsetAsync(emax1,  0, (size_t)N * HEADS * 4, stream);
    hipMemsetAsync(denom1, 0, (size_t)N * HEADS * 4, stream);
    hipMemsetAsync(out1,   0, (size_t)N * 256 * 4,   stream);
    hipMemsetAsync(emax2,  0, (size_t)N * 4,         stream);
    hipMemsetAsync(denom2, 0, (size_t)N * 4,         stream);
    hipMemsetAsync(out2,   0, (size_t)N * 64 * 4,    stream);
    hipMemsetAsync(pooled, 0, 64 * 4,                stream);

    // 1) node encoder: wave per node
    {
        int blocks = (N * 32 + 255) / 256;
        node_encoder_kernel<<<blocks, 256, 0, stream>>>(positions, degrees, enc_w,
                                                        enc_b, enc_g, enc_be, x, N);
    }
    // 2) h1 = x @ gat1_w  [N,32]x[32,256]
    {
        int tiles = ((N + 15) / 16) * (256 / 16);
        wmma_gemm_f32_kernel<<<(tiles + 7) / 8, 256, 0, stream>>>(x, gat1_w, h1,
                                                                  N, 32, 256);
    }
    // 3) identity = x @ proj_w  [N,32]x[32,64]
    {
        int tiles = ((N + 15) / 16) * (64 / 16);
        wmma_gemm_f32_kernel<<<(tiles + 7) / 8, 256, 0, stream>>>(x, proj_w, identW,
                                                                  N, 32, 64);
    }
    // 4) GAT1 attention
    {
        int total = N * HEADS;
        attn_coef_kernel<<<(total + 255) / 256, 256, 0, stream>>>(h1, gat1_as, gat1_ad,
                                                                  as1, ad1, N, HEADS);
        int et = EP * HEADS;
        edge_max_kernel<<<(et + 255) / 256, 256, 0, stream>>>(edge_idx, as1, ad1,
                                                              emax1, E_, N, HEADS);
        edge_exp_denom_kernel<<<(et + 255) / 256, 256, 0, stream>>>(edge_idx, as1, ad1,
                                                                    emax1, alpha1, denom1,
                                                                    E_, N, HEADS);
        long long at = (long long)EP * CH;
        int ablocks = (int)((at + 255) / 256);
        edge_aggregate_kernel<<<ablocks, 256, 0, stream>>>(edge_idx, alpha1, denom1,
                                                           h1, out1, E_, N, HEADS);
    }
    // 5) post1: LN + elu -> reuse h1 buffer as hpost1
    post1_kernel<<<N, 256, 0, stream>>>(out1, gat1_b, n1_g, n1_b, h1, N);

    // 6) h2 = hpost1 @ gat2_w  [N,256]x[256,64]
    {
        int tiles = ((N + 15) / 16) * (64 / 16);
        wmma_gemm_f32_kernel<<<(tiles + 7) / 8, 256, 0, stream>>>(h1, gat2_w, h2,
                                                                  N, 256, 64);
    }
    // 7) GAT2 attention (heads = 1)
    {
        attn_coef_kernel<<<(N + 255) / 256, 256, 0, stream>>>(h2, gat2_as, gat2_ad,
                                                              as2, ad2, N, 1);
        edge_max_kernel<<<(EP + 255) / 256, 256, 0, stream>>>(edge_idx, as2, ad2,
                                                              emax2, E_, N, 1);
        edge_exp_denom_kernel<<<(EP + 255) / 256, 256, 0, stream>>>(edge_idx, as2, ad2,
                                                                    emax2, alpha2, denom2,
                                                                    E_, N, 1);
        long long at = (long long)EP * CH;
        int ablocks = (int)((at + 255) / 256);
        edge_aggregate_kernel<<<ablocks, 256, 0, stream>>>(edge_idx, alpha2, denom2,
                                                           h2, out2, E_, N, 1);
    }
    // 8) post2: LN + residual + elu + pooled sum
    post2_kernel<<<N, 64, 0, stream>>>(out2, gat2_b, n2_g, n2_b, identW, proj_b,
                                       pooled, N);

    // 9) fusion head -> 256 outputs
    fusion_kernel<<<1, 256, 0, stream>>>(traffic, tr_w, tr_b, tr_g, tr_be, pooled,
                                         fu_w, fu_b, fu_g, fu_be, (float*)d_out, N);
}